// AGCN_78915729096975
// MI455X (gfx1250) — compile-verified
//
#include <hip/hip_runtime.h>

typedef __attribute__((ext_vector_type(2))) float v2f;
typedef __attribute__((ext_vector_type(4))) float v4f;
typedef __attribute__((ext_vector_type(8))) float v8f;

#define NNODES 420
#define NBATCH 64
#define KDIM   192          // CHEB_K * DIM_IN
#define WROW   640          // DIM_OUT * CLUST
#define WNODE  122880       // 3*64*64*10
#define GS_STRIDE 196
#define WT_STRIDE 18
#define CS_STRIDE 164
#define XT_STRIDE 34

static __device__ __forceinline__ v8f wmma_f32(v2f a, v2f b, v8f c) {
    // V_WMMA_F32_16X16X4_F32 : D = A(16x4) * B(4x16) + C(16x16)
    return __builtin_amdgcn_wmma_f32_16x16x4_f32(
        /*neg_a=*/false, a, /*neg_b=*/false, b,
        /*c_mod=*/(short)0, c, /*reuse_a=*/false, /*reuse_b=*/false);
}

// ---------------------------------------------------------------------------
// Kernel A: T2 = 2*L@L - I   (420x420, one wave per 16x16 C-tile)
// ---------------------------------------------------------------------------
__global__ __launch_bounds__(32) void cheb2_kernel(const float* __restrict__ L,
                                                   float* __restrict__ T2) {
    const int mt = blockIdx.x / 27, nt = blockIdx.x % 27;
    const int lane = threadIdx.x;
    const int lm = lane & 15, lh = lane >> 4;
    int rA = mt * 16 + lm; if (rA > 419) rA = 419;   // clamp: junk rows never stored
    int cB = nt * 16 + lm; if (cB > 419) cB = 419;

    v8f acc = {0.f, 0.f, 0.f, 0.f, 0.f, 0.f, 0.f, 0.f};
    for (int k0 = 0; k0 < 420; k0 += 4) {
        const int k = k0 + 2 * lh;
        v2f a = *(const v2f*)(L + rA * 420 + k);     // A = L, row-major, b64
        v2f b;
        b.x = L[k * 420 + cB];                       // B = L (K x N)
        b.y = L[(k + 1) * 420 + cB];
        acc = wmma_f32(a, b, acc);
    }
    const int gn = nt * 16 + lm;
    const int gm0 = mt * 16 + 8 * lh;
    if (gn < 420) {
#pragma unroll
        for (int v = 0; v < 8; ++v) {
            const int gm = gm0 + v;
            if (gm < 420) {
                const float eye = (gm == gn) ? 1.f : 0.f;
                T2[gm * 420 + gn] = 2.f * acc[v] - eye;
            }
        }
    }
}

// ---------------------------------------------------------------------------
// Kernel B: x_g.  For each batch b: out_k[n,c] = S_k[n,:] @ x[b,:,:], k=1,2
// plus k=0 identity copy.  Output layout xg[n][b][192] (k*64+c innermost).
// Grid: 64 batches x 7 groups of 4 M-tiles; 128 threads (4 waves).
// ---------------------------------------------------------------------------
__global__ __launch_bounds__(128) void xg_kernel(const float* __restrict__ x,
                                                 const float* __restrict__ L,
                                                 const float* __restrict__ T2,
                                                 float* __restrict__ xg) {
    __shared__ float Xt[64 * XT_STRIDE];             // x[b] K-chunk, transposed [c][mloc]

    const int b  = blockIdx.x / 7;
    const int mg = blockIdx.x % 7;
    const int tid = threadIdx.x;
    const int w = tid >> 5, lane = tid & 31;
    const int lm = lane & 15, lh = lane >> 4;
    const int m0 = (mg * 4 + w) * 16;                // this wave's 16 output rows (nodes)
    int rA = m0 + lm; if (rA > 419) rA = 419;
    const float* __restrict__ xb = x + (size_t)b * (NNODES * 64);

    v8f accL[4], accT[4];
    const v8f vzero = {0.f, 0.f, 0.f, 0.f, 0.f, 0.f, 0.f, 0.f};
#pragma unroll
    for (int t = 0; t < 4; ++t) { accL[t] = vzero; accT[t] = vzero; }

    for (int ch = 0; ch < 14; ++ch) {                // 14 chunks of 32 K-rows (420 padded)
        // stage x[b][ch*32 .. ch*32+31][0..63] transposed into LDS (b128 reads)
#pragma unroll
        for (int i = 0; i < 4; ++i) {
            const int e4 = tid * 4 + i * 512;        // 2048 floats per chunk
            const int r = e4 >> 6, c = e4 & 63;      // c is a multiple of 4
            const int m = ch * 32 + r;
            const int mc = (m > 419) ? 419 : m;
            v4f v = *(const v4f*)(&xb[mc * 64 + c]);
            if (m > 419) { v4f z = {0.f, 0.f, 0.f, 0.f}; v = z; }
#pragma unroll
            for (int q = 0; q < 4; ++q)
                Xt[(c + q) * XT_STRIDE + r] = v[q];
        }
        __syncthreads();
#pragma unroll
        for (int ks = 0; ks < 8; ++ks) {
            int k = ch * 32 + ks * 4 + 2 * lh;
            if (k > 418) k = 418;                    // junk A * zero B = 0
            v2f aL = *(const v2f*)(L  + rA * 420 + k);
            v2f aT = *(const v2f*)(T2 + rA * 420 + k);
#pragma unroll
            for (int t = 0; t < 4; ++t) {
                const int c = t * 16 + lm;
                v2f bx = *(const v2f*)(&Xt[c * XT_STRIDE + ks * 4 + 2 * lh]);
                accL[t] = wmma_f32(aL, bx, accL[t]);
                accT[t] = wmma_f32(aT, bx, accT[t]);
            }
        }
        __syncthreads();
    }
    // epilogue: xg[n][b][0:64]=x, [64:128]=L@x, [128:192]=T2@x
#pragma unroll
    for (int v = 0; v < 8; ++v) {
        const int n = m0 + 8 * lh + v;
        if (n < 420) {
            float* __restrict__ dst = xg + (size_t)n * (NBATCH * KDIM) + (size_t)b * KDIM;
#pragma unroll
            for (int t = 0; t < 4; ++t) {
                const int c = t * 16 + lm;
                dst[c]       = xb[n * 64 + c];
                dst[64 + c]  = accL[t][v];
                dst[128 + c] = accT[t][v];
            }
        }
    }
}

// ---------------------------------------------------------------------------
// Kernel C: per (node, 16-o block): C[64,160] = G[64,192] @ W2[192,160],
// then bias + max over 10 clusters -> out[b][n][o].
// Grid: 420*4 blocks, 128 threads (4 waves, one M-tile each, 10 N-tiles).
// This kernel carries the 206 MB weight stream -> b128 everywhere on globals.
// ---------------------------------------------------------------------------
__global__ __launch_bounds__(128) void gconv_kernel(const float* __restrict__ xg,
                                                    const float* __restrict__ Wp,
                                                    const float* __restrict__ bias,
                                                    float* __restrict__ out) {
    __shared__ float Gs[64 * GS_STRIDE];             // G tile (reused as C spill)
    __shared__ float Wt[2][160 * WT_STRIDE];         // double-buffered W chunk, [c][k]

    const int n  = blockIdx.x >> 2;
    const int o0 = (blockIdx.x & 3) * 16;
    const int tid = threadIdx.x;
    const int w = tid >> 5, lane = tid & 31;
    const int lm = lane & 15, lh = lane >> 4;
    const int m0 = w * 16;

    const float* __restrict__ xgn = xg + (size_t)n * (NBATCH * KDIM);
    const float* __restrict__ Wn  = Wp + (size_t)n * WNODE + o0 * 10;

    // stage G: [64 rows][192] -> padded stride 196; pure b128 (784 = 49*16)
#pragma unroll
    for (int i = 0; i < 24; ++i) {
        const int e4 = tid * 4 + i * 512;            // 12288 floats total
        const int m = e4 / KDIM, k = e4 - m * KDIM;  // k is a multiple of 4
        *(v4f*)(&Gs[m * GS_STRIDE + k]) = *(const v4f*)(&xgn[e4]);
    }
    // stage W chunk 0, b128 loads, scatter-transpose to [c][k]
    v4f pf[5];
#pragma unroll
    for (int i = 0; i < 5; ++i) {
        const int e4 = tid * 4 + i * 512;            // 2560 floats per chunk
        const int r = e4 / 160, c = e4 - r * 160;
        pf[i] = *(const v4f*)(Wn + (size_t)r * WROW + c);
    }
#pragma unroll
    for (int i = 0; i < 5; ++i) {
        const int e4 = tid * 4 + i * 512;
        const int r = e4 / 160, c = e4 - r * 160;
#pragma unroll
        for (int q = 0; q < 4; ++q)
            Wt[0][(c + q) * WT_STRIDE + r] = pf[i][q];
    }
    __syncthreads();

    v8f acc[10];
    const v8f vzero = {0.f, 0.f, 0.f, 0.f, 0.f, 0.f, 0.f, 0.f};
#pragma unroll
    for (int t = 0; t < 10; ++t) acc[t] = vzero;

    for (int j = 0; j < 12; ++j) {                   // 12 K-chunks of 16
        if (j < 10) {                                // L2 prefetch, two chunks ahead
            __builtin_prefetch(Wn + (size_t)((j + 2) * 16 + (tid >> 3)) * WROW +
                                   (tid & 7) * 20, 0, 0);
        }
        if (j < 11) {                                // register prefetch of next chunk
#pragma unroll
            for (int i = 0; i < 5; ++i) {
                const int e4 = tid * 4 + i * 512;
                const int r = e4 / 160, c = e4 - r * 160;
                pf[i] = *(const v4f*)(Wn + (size_t)((j + 1) * 16 + r) * WROW + c);
            }
        }
        const int buf = j & 1;
#pragma unroll
        for (int ks = 0; ks < 4; ++ks) {
            const int kk = j * 16 + ks * 4 + 2 * lh;
            v2f a = *(const v2f*)(&Gs[(m0 + lm) * GS_STRIDE + kk]);
#pragma unroll
            for (int t = 0; t < 10; ++t) {
                v2f bv = *(const v2f*)(&Wt[buf][(t * 16 + lm) * WT_STRIDE + ks * 4 + 2 * lh]);
                acc[t] = wmma_f32(a, bv, acc[t]);
            }
        }
        if (j < 11) {
#pragma unroll
            for (int i = 0; i < 5; ++i) {
                const int e4 = tid * 4 + i * 512;
                const int r = e4 / 160, c = e4 - r * 160;
#pragma unroll
                for (int q = 0; q < 4; ++q)
                    Wt[(j + 1) & 1][(c + q) * WT_STRIDE + r] = pf[i][q];
            }
        }
        __syncthreads();
    }

    // spill C tiles to LDS (reuse Gs region), then bias + max over clusters
    float* Cs = Gs;
#pragma unroll
    for (int t = 0; t < 10; ++t) {
#pragma unroll
        for (int v = 0; v < 8; ++v) {
            const int m = m0 + 8 * lh + v;
            Cs[m * CS_STRIDE + t * 16 + lm] = acc[t][v];
        }
    }
    __syncthreads();

    const int o = tid & 15, mgrp = tid >> 4;
    float bv[10];
#pragma unroll
    for (int cl = 0; cl < 10; ++cl)
        bv[cl] = bias[(size_t)n * WROW + (o0 + o) * 10 + cl];
#pragma unroll
    for (int it = 0; it < 8; ++it) {
        const int m = it * 8 + mgrp;                 // m == batch index
        float mx = -3.402823466e38f;
#pragma unroll
        for (int cl = 0; cl < 10; ++cl)
            mx = fmaxf(mx, Cs[m * CS_STRIDE + o * 10 + cl] + bv[cl]);
        out[(size_t)m * (NNODES * 64) + n * 64 + o0 + o] = mx;
    }
}

// ---------------------------------------------------------------------------
extern "C" void kernel_launch(void* const* d_in, const int* in_sizes, int n_in,
                              void* d_out, int out_size, void* d_ws, size_t ws_size,
                              hipStream_t stream) {
    (void)in_sizes; (void)n_in; (void)out_size; (void)ws_size;
    const float* x    = (const float*)d_in[0];
    // d_in[1] = node_embeddings: unused by the reference computation
    const float* L    = (const float*)d_in[2];
    const float* Wp   = (const float*)d_in[3];
    const float* bias = (const float*)d_in[4];
    float* out = (float*)d_out;

    float* T2 = (float*)d_ws;                 // 420*420
    float* xg = T2 + NNODES * NNODES;         // 420*64*192  (~21.4 MB total ws)

    cheb2_kernel<<<27 * 27, 32, 0, stream>>>(L, T2);
    xg_kernel<<<NBATCH * 7, 128, 0, stream>>>(x, L, T2, xg);
    gconv_kernel<<<NNODES * 4, 128, 0, stream>>>(xg, Wp, bias, out);
}